// PatchedGroup_42348377538666
// MI455X (gfx1250) — compile-verified
//
#include <hip/hip_runtime.h>
#include <hip/hip_bf16.h>
#include <cstdint>

#define BATCH  16
#define NPTS   16384
#define CHAN   6
#define NGROUP 512
#define GSIZE  32

#define GT     16                 // centers per KNN block (one WMMA tile of M)
#define CHUNK  1024               // points per chunk
#define SPAD   (CHUNK + 16)       // padded LDS row stride (bank spread)
#define NCHUNK (NPTS / CHUNK)     // 16
#define NCAND  (NCHUNK * GSIZE)   // 512 candidates per center

typedef __attribute__((ext_vector_type(2))) float v2f;
typedef __attribute__((ext_vector_type(8))) float v8f;
typedef __attribute__((ext_vector_type(4))) int   v4i;

// CDNA5 async global->LDS path (ASYNCcnt-tracked), guarded so the file
// compiles on toolchains where the builtin is absent.
#if defined(__has_builtin)
#  if __has_builtin(__builtin_amdgcn_global_load_async_to_lds_b128) && \
      __has_builtin(__builtin_amdgcn_s_wait_asynccnt)
#    define PG_ASYNC_LDS 1
#  endif
#endif
#ifndef PG_ASYNC_LDS
#  define PG_ASYNC_LDS 0
#endif

#if PG_ASYNC_LDS
// Exact param types per hipcc diagnostic: v4i in AS1 (global) / AS3 (LDS).
typedef __attribute__((address_space(1))) v4i g_v4i;
typedef __attribute__((address_space(3))) v4i l_v4i;
#endif

// ---------------- Kernel 0: AoS [B,N,6] -> SoA px/py/pz + p2 ----------------
__global__ void pg_prep(const float* __restrict__ xyz,
                        float* __restrict__ px, float* __restrict__ py,
                        float* __restrict__ pz, float* __restrict__ p2) {
  int i = blockIdx.x * blockDim.x + threadIdx.x;
  if (i >= BATCH * NPTS) return;
  size_t base = (size_t)i * CHAN;
  float x = xyz[base + 0], y = xyz[base + 1], z = xyz[base + 2];
  px[i] = x; py[i] = y; pz[i] = z; p2[i] = x * x + y * y + z * z;
}

// ---------------- Kernel 1: FPS, one block per batch, points in LDS ---------
__global__ void __launch_bounds__(1024) pg_fps(
    const float* __restrict__ px, const float* __restrict__ py,
    const float* __restrict__ pz,
    float* __restrict__ out_center,
    float* __restrict__ csx, float* __restrict__ csy,
    float* __restrict__ csz, float* __restrict__ cs2) {
  extern __shared__ float fsm[];           // 3 * NPTS floats = 192 KB
  float* lx = fsm;
  float* ly = lx + NPTS;
  float* lz = ly + NPTS;

  int b   = blockIdx.x;
  int tid = threadIdx.x;
  int lane = tid & 31, wid = tid >> 5;

  __shared__ float rv[32];
  __shared__ int   ri[32];
  __shared__ float sc[3];
  __shared__ int   sfar;

  const float* bx = px + (size_t)b * NPTS;
  const float* by = py + (size_t)b * NPTS;
  const float* bz = pz + (size_t)b * NPTS;

#if PG_ASYNC_LDS
  // stage points into LDS with GLOBAL_LOAD_ASYNC_TO_LDS_B128 (16B per op)
  for (int i = tid * 4; i < NPTS; i += 4096) {
    __builtin_amdgcn_global_load_async_to_lds_b128((g_v4i*)(bx + i), (l_v4i*)(lx + i), 0, 0);
    __builtin_amdgcn_global_load_async_to_lds_b128((g_v4i*)(by + i), (l_v4i*)(ly + i), 0, 0);
    __builtin_amdgcn_global_load_async_to_lds_b128((g_v4i*)(bz + i), (l_v4i*)(lz + i), 0, 0);
  }
  __builtin_amdgcn_s_wait_asynccnt(0);     // per-wave; barrier below publishes
#else
  for (int i = tid; i < NPTS; i += 1024) { lx[i] = bx[i]; ly[i] = by[i]; lz[i] = bz[i]; }
#endif
  if (tid == 0) sfar = 0;

  float md[16];
#pragma unroll
  for (int i = 0; i < 16; ++i) md[i] = 1e10f;   // matches reference init

  for (int s = 0; s < NGROUP; ++s) {
    __syncthreads();
    if (tid == 0) {
      int f = sfar;
      float x = lx[f], y = ly[f], z = lz[f];
      sc[0] = x; sc[1] = y; sc[2] = z;
      size_t o = (size_t)b * NGROUP + s;
      out_center[o * 3 + 0] = x; out_center[o * 3 + 1] = y; out_center[o * 3 + 2] = z;
      csx[o] = x; csy[o] = y; csz[o] = z; cs2[o] = x * x + y * y + z * z;
    }
    __syncthreads();
    float cx = sc[0], cy = sc[1], cz = sc[2];
    float bv = -1.0f; int bi = 0;
#pragma unroll
    for (int i = 0; i < 16; ++i) {
      int idx = tid + i * 1024;
      float dx = lx[idx] - cx, dy = ly[idx] - cy, dz = lz[idx] - cz;
      float d2 = dx * dx + dy * dy + dz * dz;
      md[i] = fminf(md[i], d2);
      if (md[i] > bv) { bv = md[i]; bi = idx; }   // strict > keeps lowest idx
    }
    for (int off = 16; off; off >>= 1) {          // wave32 argmax, tie -> lower idx
      float ov = __shfl_down(bv, off, 32);
      int   oi = __shfl_down(bi, off, 32);
      if (ov > bv || (ov == bv && oi < bi)) { bv = ov; bi = oi; }
    }
    if (lane == 0) { rv[wid] = bv; ri[wid] = bi; }
    __syncthreads();
    if (wid == 0) {
      bv = rv[lane]; bi = ri[lane];
      for (int off = 16; off; off >>= 1) {
        float ov = __shfl_down(bv, off, 32);
        int   oi = __shfl_down(bi, off, 32);
        if (ov > bv || (ov == bv && oi < bi)) { bv = ov; bi = oi; }
      }
      if (lane == 0) sfar = bi;
    }
  }
}

// ---------------- Kernel 2: KNN via V_WMMA_F32_16X16X4_F32 + top-32 + gather
// d2 = C + A*B with A = -2*center (16x4, K padded), B = point (4x16),
// C[m][n] = c2[m] + p2[n]  =>  D is exactly the squared distance tile.
__global__ void __launch_bounds__(512) pg_knn(
    const float* __restrict__ px, const float* __restrict__ py,
    const float* __restrict__ pz, const float* __restrict__ p2,
    const float* __restrict__ csx, const float* __restrict__ csy,
    const float* __restrict__ csz, const float* __restrict__ cs2,
    const float* __restrict__ xyz,
    float* __restrict__ out_neigh, float* __restrict__ out_idx) {
  extern __shared__ float smem[];
  float* sd2   = smem;                               // GT*SPAD
  float* candD = sd2 + GT * SPAD;                    // GT*NCAND
  int*   candI = (int*)(candD + GT * NCAND);         // GT*NCAND
  float* scx   = (float*)(candI + GT * NCAND);
  float* scy   = scx + GT;
  float* scz   = scy + GT;
  float* sc2   = scz + GT;
  int*   finalI = (int*)(sc2 + GT);                  // GT*GSIZE

  int b  = blockIdx.x / (NGROUP / GT);
  int g0 = (blockIdx.x % (NGROUP / GT)) * GT;
  int tid = threadIdx.x, lane = tid & 31, wid = tid >> 5;

  if (tid < GT) {
    size_t o = (size_t)b * NGROUP + g0 + tid;
    scx[tid] = csx[o]; scy[tid] = csy[o]; scz[tid] = csz[o]; sc2[tid] = cs2[o];
  }
  __syncthreads();

  const float* bx  = px + (size_t)b * NPTS;
  const float* by  = py + (size_t)b * NPTS;
  const float* bz  = pz + (size_t)b * NPTS;
  const float* bp2 = p2 + (size_t)b * NPTS;

  int hi = lane >> 4;        // half-wave select (K 0,1 vs K 2,3)
  int m  = lane & 15;

  // Branchless per-lane source select: one pointer cndmask outside the loop
  // instead of exec-mask branches around every tile's loads.
  const float* src0 = hi ? bz : bx;

  // A-matrix (16x4 f32): lanes 0-15 hold K0,K1; lanes 16-31 hold K2,K3
  v2f A;
  A.x = -2.0f * (hi ? scz[m] : scx[m]);
  A.y = hi ? 0.0f : (-2.0f * scy[m]);
  float c2r[8];
#pragma unroll
  for (int r = 0; r < 8; ++r) c2r[r] = sc2[r + hi * 8];  // D rows for this lane

  const float INF = 3.0e38f;

  for (int ch = 0; ch < NCHUNK; ++ch) {
    int base = ch * CHUNK;
#pragma unroll
    for (int t4 = 0; t4 < 4; ++t4) {
      int tile = wid * 4 + t4;
      int col  = tile * 16 + m;
      int pidx = base + col;
      if (ch + 1 < NCHUNK) __builtin_prefetch(&src0[pidx + CHUNK], 0, 1);
      v2f Bm;                                   // B-matrix (4x16 f32), column = m
      Bm.x = src0[pidx];                        // unconditional load via selected ptr
      float ty = by[pidx];                      // unconditional load + cndmask
      Bm.y = hi ? 0.0f : ty;
      float pp = bp2[pidx];
      v8f C;
#pragma unroll
      for (int r = 0; r < 8; ++r) C[r] = c2r[r] + pp;
      v8f D = __builtin_amdgcn_wmma_f32_16x16x4_f32(
          /*neg_a=*/false, A, /*neg_b=*/false, Bm,
          /*c_mod=*/(short)0, C, /*reuse_a=*/false, /*reuse_b=*/false);
#pragma unroll
      for (int r = 0; r < 8; ++r) sd2[(r + hi * 8) * SPAD + col] = D[r];
    }
    __syncthreads();

    // chunk top-32 per center: wave `wid` owns center `wid`, values in regs
    float v[32];
#pragma unroll
    for (int j = 0; j < 32; ++j) v[j] = sd2[wid * SPAD + j * 32 + lane];
    for (int r = 0; r < GSIZE; ++r) {
      float bv = INF; int bp = 0;
#pragma unroll
      for (int j = 0; j < 32; ++j)
        if (v[j] < bv) { bv = v[j]; bp = j * 32 + lane; }
      for (int off = 16; off; off >>= 1) {      // wave argmin, tie -> lower pos
        float ov = __shfl_xor(bv, off, 32);
        int   op = __shfl_xor(bp, off, 32);
        if (ov < bv || (ov == bv && op < bp)) { bv = ov; bp = op; }
      }
#pragma unroll
      for (int j = 0; j < 32; ++j)              // invalidate winner (no dyn idx)
        if (j * 32 + lane == bp) v[j] = INF;
      if (lane == 0) {
        candD[wid * NCAND + ch * GSIZE + r] = bv;
        candI[wid * NCAND + ch * GSIZE + r] = base + bp;
      }
    }
    __syncthreads();
  }

  // final exact top-32 of the 512 candidates per center (ascending, tie->idx)
  float fv[16]; int fi[16];
#pragma unroll
  for (int j = 0; j < 16; ++j) {
    int c = j * 32 + lane;
    fv[j] = candD[wid * NCAND + c];
    fi[j] = candI[wid * NCAND + c];
  }
  for (int r = 0; r < GSIZE; ++r) {
    float bv = INF; int bpt = 0x7fffffff; int bcode = -1;
#pragma unroll
    for (int j = 0; j < 16; ++j)
      if (fv[j] < bv || (fv[j] == bv && fi[j] < bpt)) {
        bv = fv[j]; bpt = fi[j]; bcode = j * 32 + lane;
      }
    for (int off = 16; off; off >>= 1) {
      float ov = __shfl_xor(bv, off, 32);
      int   opt = __shfl_xor(bpt, off, 32);
      int   oc  = __shfl_xor(bcode, off, 32);
      if (ov < bv || (ov == bv && opt < bpt)) { bv = ov; bpt = opt; bcode = oc; }
    }
#pragma unroll
    for (int j = 0; j < 16; ++j)
      if (j * 32 + lane == bcode) fv[j] = INF;
    if (lane == 0) {
      finalI[wid * GSIZE + r] = bpt;
      out_idx[((size_t)b * NGROUP + g0 + wid) * GSIZE + r] = (float)bpt;
    }
  }
  __syncthreads();

  // gather: thread -> (center c, neighbor j)
  {
    int c = tid >> 5, j = tid & 31;
    int pti = finalI[c * GSIZE + j];
    const float* src = xyz + ((size_t)b * NPTS + pti) * CHAN;
    size_t o = (((size_t)b * NGROUP + g0 + c) * GSIZE + j) * CHAN;
    out_neigh[o + 0] = src[0] - scx[c];
    out_neigh[o + 1] = src[1] - scy[c];
    out_neigh[o + 2] = src[2] - scz[c];
    out_neigh[o + 3] = src[3];
    out_neigh[o + 4] = src[4];
    out_neigh[o + 5] = src[5];
  }
}

extern "C" void kernel_launch(void* const* d_in, const int* in_sizes, int n_in,
                              void* d_out, int out_size, void* d_ws, size_t ws_size,
                              hipStream_t stream) {
  const float* xyz = (const float*)d_in[0];
  float* ws = (float*)d_ws;
  const size_t BN = (size_t)BATCH * NPTS;
  const size_t BG = (size_t)BATCH * NGROUP;

  float* px  = ws;
  float* py  = px + BN;
  float* pz  = py + BN;
  float* p2  = pz + BN;
  float* csx = p2 + BN;
  float* csy = csx + BG;
  float* csz = csy + BG;
  float* cs2 = csz + BG;

  float* out        = (float*)d_out;
  float* out_neigh  = out;                                            // [B,G,32,6]
  float* out_center = out + (size_t)BATCH * NGROUP * GSIZE * CHAN;    // [B,G,3]
  float* out_idx    = out_center + (size_t)BATCH * NGROUP * 3;        // [B,G,32]

  pg_prep<<<(BATCH * NPTS + 255) / 256, 256, 0, stream>>>(xyz, px, py, pz, p2);

  size_t fps_lds = (size_t)3 * NPTS * sizeof(float);                  // 192 KB
  pg_fps<<<BATCH, 1024, fps_lds, stream>>>(px, py, pz, out_center,
                                           csx, csy, csz, cs2);

  size_t knn_lds = (size_t)GT * SPAD * sizeof(float)                  // d2 tile
                 + (size_t)GT * NCAND * (sizeof(float) + sizeof(int)) // candidates
                 + 4 * GT * sizeof(float)                             // centers
                 + (size_t)GT * GSIZE * sizeof(int);                  // final idx
  pg_knn<<<BATCH * (NGROUP / GT), 512, knn_lds, stream>>>(
      px, py, pz, p2, csx, csy, csz, cs2, xyz, out_neigh, out_idx);
}